// GCNAggregator_39797166964866
// MI455X (gfx1250) — compile-verified
//
#include <hip/hip_runtime.h>

typedef __attribute__((ext_vector_type(2))) float v2f;
typedef __attribute__((ext_vector_type(4))) float v4f;
typedef __attribute__((ext_vector_type(8))) float v8f;

#define EMB 64

// ---------------------------------------------------------------------------
// Zero the output (atomic scatter target must start at 0; harness poisons it).
// ---------------------------------------------------------------------------
__global__ __launch_bounds__(256)
void zero_kernel(v4f* __restrict__ out, int n4) {
    int i = blockIdx.x * blockDim.x + threadIdx.x;
    if (i < n4) {
        v4f z = {0.0f, 0.0f, 0.0f, 0.0f};
        out[i] = z;
    }
}

// ---------------------------------------------------------------------------
// FW = feature @ W^T   using V_WMMA_F32_16X16X4_F32.
// feature: [nrows, 64] row-major, W: [64, 64] row-major.
// FW[i, j] = sum_k feature[i, k] * W[j, k]
// One wave computes a 16-row x 64-col strip: 4 C tiles, 16 K-steps of K=4.
//
// B fragments are staged in LDS as K-pair-interleaved float2 so each lane's
// {B[k][n], B[k+1][n]} is one aligned ds_load_b64 straight into the even-
// aligned VGPR pair WMMA consumes (no v_mov reshuffles).
//   ldsP[(k>>1)*64 + n] = { W[n][k], W[n][k+1] }
// ---------------------------------------------------------------------------
__global__ __launch_bounds__(256)
void fw_gemm_kernel(const float* __restrict__ F, const float* __restrict__ W,
                    float* __restrict__ FW, int nrows) {
    __shared__ v2f ldsP[(EMB / 2) * EMB];   // 32 k-pairs x 64 n  = 16 KB
    for (int idx = threadIdx.x; idx < (EMB / 2) * EMB; idx += blockDim.x) {
        int k2 = idx >> 6;      // k pair index
        int n  = idx & 63;      // output column (row of W)
        v2f t;
        t.x = W[n * EMB + 2 * k2];
        t.y = W[n * EMB + 2 * k2 + 1];
        ldsP[idx] = t;          // idx == k2*64 + n
    }
    __syncthreads();

    const int wave  = threadIdx.x >> 5;
    const int lane  = threadIdx.x & 31;
    const int lm    = lane & 15;                 // matrix row/col within tile
    const int khalf = (lane >> 4) << 1;          // 0 for lanes 0-15, 2 for 16-31
    const int rowBase = blockIdx.x * 128 + wave * 16;
    if (rowBase + 16 > nrows) return;            // wave-uniform: EXEC stays all-1s

    const float* __restrict__ arow = F + (size_t)(rowBase + lm) * EMB + khalf;

    v8f acc0 = {}, acc1 = {}, acc2 = {}, acc3 = {};

    #pragma unroll
    for (int kk = 0; kk < EMB; kk += 4) {
        // A fragment: lane holds F[rowBase+lm, kk+khalf .. kk+khalf+1]
        v2f a;
        a.x = arow[kk];
        a.y = arow[kk + 1];

        // B fragments: one ds_load_b64 each
        const v2f* __restrict__ bp = &ldsP[((kk + khalf) >> 1) * EMB + lm];
        v2f b0 = bp[0];
        v2f b1 = bp[16];
        v2f b2 = bp[32];
        v2f b3 = bp[48];

        acc0 = __builtin_amdgcn_wmma_f32_16x16x4_f32(false, a, false, b0, (short)0, acc0, false, false);
        acc1 = __builtin_amdgcn_wmma_f32_16x16x4_f32(false, a, false, b1, (short)0, acc1, false, false);
        acc2 = __builtin_amdgcn_wmma_f32_16x16x4_f32(false, a, false, b2, (short)0, acc2, false, false);
        acc3 = __builtin_amdgcn_wmma_f32_16x16x4_f32(false, a, false, b3, (short)0, acc3, false, false);
    }

    // C/D layout: VGPR r -> row r (lanes 0-15) or row r+8 (lanes 16-31), N = lane%16
    const int rOff = rowBase + ((lane >> 4) << 3);
    #pragma unroll
    for (int r = 0; r < 8; ++r) {
        float* __restrict__ dst = FW + (size_t)(rOff + r) * EMB + lm;
        dst[0]  = acc0[r];
        dst[16] = acc1[r];
        dst[32] = acc2[r];
        dst[48] = acc3[r];
    }
}

// ---------------------------------------------------------------------------
// out[row] += val * FW[col]   (COO scatter-add). 16 threads per edge, each
// handling a float4 chunk of the 64-wide embedding. FW and out tiles live in
// the 192 MB L2, so the random gathers/atomics stay on-chip.
// ---------------------------------------------------------------------------
__global__ __launch_bounds__(256)
void scatter_kernel(const int* __restrict__ rows, const int* __restrict__ cols,
                    const float* __restrict__ vals, const float* __restrict__ FW,
                    float* __restrict__ out, int nedges) {
    long long idx = (long long)blockIdx.x * blockDim.x + threadIdx.x;
    int e = (int)(idx >> 4);
    if (e >= nedges) return;
    int c = ((int)idx & 15) << 2;

    int row = rows[e];
    int col = cols[e];
    float v = vals[e];

    v4f m = *(const v4f*)(FW + (size_t)col * EMB + c);
    float* dst = out + (size_t)row * EMB + c;
    atomicAdd(dst + 0, v * m.x);
    atomicAdd(dst + 1, v * m.y);
    atomicAdd(dst + 2, v * m.z);
    atomicAdd(dst + 3, v * m.w);
}

// ---------------------------------------------------------------------------
extern "C" void kernel_launch(void* const* d_in, const int* in_sizes, int n_in,
                              void* d_out, int out_size, void* d_ws, size_t ws_size,
                              hipStream_t stream) {
    const int*   rows = (const int*)d_in[0];
    const int*   cols = (const int*)d_in[1];
    const float* vals = (const float*)d_in[2];
    const float* F    = (const float*)d_in[3];
    const float* W    = (const float*)d_in[4];
    float* out = (float*)d_out;
    float* FW  = (float*)d_ws;   // [N, 64] scratch = feature @ W^T

    const int E = in_sizes[0];
    const int N = in_sizes[3] / EMB;

    // 1) out = 0
    int n4 = out_size >> 2;
    zero_kernel<<<(n4 + 255) / 256, 256, 0, stream>>>((v4f*)out, n4);

    // 2) FW = F @ W^T  (WMMA; 8 waves/block, 128 rows per block)
    int gblocks = (N + 127) / 128;
    fw_gemm_kernel<<<gblocks, 256, 0, stream>>>(F, W, FW, N);

    // 3) out[row] += val * FW[col]
    long long tot = (long long)E * 16;
    scatter_kernel<<<(int)((tot + 255) / 256), 256, 0, stream>>>(rows, cols, vals, FW, out, E);
}